// SpectralRegularizer_65927747994033
// MI455X (gfx1250) — compile-verified
//
#include <hip/hip_runtime.h>
#include <hip/hip_bf16.h>

// SpectralRegularizer, mathematically collapsed (see round-0 analysis):
//   All k_mag fall into spectrum bin 1 (bins are linspace(0,64,64), k_mag<=0.866).
//   loss = 0.01 * mean(expected_n^2)            (constant)
//        + 0.001/63 * E_rfft / (count*b)
//   E_rfft = 0.5*(128^3 * sum(x^2) + 128^2 * sum_cols[(sum_z x)^2 + (sum_z (-1)^z x)^2])
// => single streaming pass over 100MB input; memory-bound (~4.3us at 23.3 TB/s).
// Column sums on the matrix pipe via V_WMMA_F32_16X16X4_F32; staging via
// GLOBAL_LOAD_ASYNC_TO_LDS_B128 (ASYNCcnt-tracked global->LDS DMA).

typedef __attribute__((ext_vector_type(2))) float v2f;
typedef __attribute__((ext_vector_type(8))) float v8f;
typedef __attribute__((ext_vector_type(4))) int   v4i;

// Pointer-to-AS1 / pointer-to-AS3 element types for the async builtin
// (diagnostic showed param type: 'int __attribute__((vector_size(16))) __device__ *').
typedef __attribute__((address_space(1))) v4i* gptr_v4i;
typedef __attribute__((address_space(3))) v4i* lptr_v4i;

#ifndef __has_builtin
#define __has_builtin(x) 0
#endif

#if __has_builtin(__builtin_amdgcn_wmma_f32_16x16x4_f32)
#define USE_WMMA_F32X4 1
#else
#define USE_WMMA_F32X4 0
#endif

#if __has_builtin(__builtin_amdgcn_global_load_async_to_lds_b128)
#define HAVE_ASYNC_LDS 1
#else
#define HAVE_ASYNC_LDS 0
#endif

#define NBLK_A 512
#define WAVES 4
#define BLOCK_A (WAVES * 32)
#define TILES_TOTAL 12288            // 4*3*128*128 columns / 16 cols per tile
#define SLOTS (NBLK_A * WAVES)       // 2048 wave slots
#define TILES_PER_WAVE (TILES_TOTAL / SLOTS)  // 6
#define COL_PAD 132                  // 128 dwords + 4 pad -> conflict-free b64 reads

__global__ __launch_bounds__(BLOCK_A)
void spectral_partial_kernel(const float* __restrict__ x, float* __restrict__ part) {
  __shared__ float lds[WAVES][16 * COL_PAD];
  __shared__ float wA[WAVES], wB[WAVES];

  const int lane = threadIdx.x & 31;
  const int wave = threadIdx.x >> 5;
  const int slot = blockIdx.x * WAVES + wave;   // 0..2047
  const int col  = lane & 15;                   // A-matrix M = lane&15
  const int zb   = (lane >> 4) * 2;             // lanes 16-31 hold K=2,3

  float totA = 0.0f;   // sum of x^2 over this wave's data
  float totB = 0.0f;   // sum of (s1^2 + sa^2) over this wave's columns (x16 in WMMA path)

  const v2f bones = {1.0f, 1.0f};   // B: all-ones column(s)  -> s1 = sum_z x
  const v2f balt  = {1.0f, -1.0f};  // B: (+,-,+,-) over K    -> sa = sum_z (-1)^z x

  for (int i = 0; i < TILES_PER_WAVE; ++i) {
    const size_t tile = (size_t)slot + (size_t)i * SLOTS;      // 0..12287
    const float* src = x + tile * 2048;                        // 16 cols * 128 z, contiguous

#if HAVE_ASYNC_LDS
    // Direct global->LDS DMA (ASYNCcnt), 512B per instruction, padded col stride.
    {
      char* lbase = (char*)&lds[wave][0];
      const char* gbase = (const char*)src;
      #pragma unroll
      for (int it = 0; it < 16; ++it) {
        __builtin_amdgcn_global_load_async_to_lds_b128(
            (gptr_v4i)(gbase + it * 512 + lane * 16),
            (lptr_v4i)(lbase + it * (COL_PAD * 4) + lane * 16),
            0, 0);
      }
    }
#if __has_builtin(__builtin_amdgcn_s_wait_asynccnt)
    __builtin_amdgcn_s_wait_asynccnt(0);
#else
    asm volatile("s_wait_asynccnt 0" ::: "memory");
#endif
#else
    // Fallback: stage through VGPRs, coalesced b128.
    #pragma unroll
    for (int it = 0; it < 16; ++it) {
      const float4 v = *(const float4*)(src + it * 128 + lane * 4);
      *(float4*)&lds[wave][it * COL_PAD + lane * 4] = v;
    }
#endif
    __syncthreads();

    const float* cptr = &lds[wave][col * COL_PAD + zb];

#if USE_WMMA_F32X4
    v8f acc1 = {0.f, 0.f, 0.f, 0.f, 0.f, 0.f, 0.f, 0.f};
    v8f acca = {0.f, 0.f, 0.f, 0.f, 0.f, 0.f, 0.f, 0.f};
    #pragma unroll
    for (int k = 0; k < 32; ++k) {
      const v2f a = *(const v2f*)(cptr + 4 * k);   // A frag: (z, z+1) for this lane
      totA += a.x * a.x + a.y * a.y;
      acc1 = __builtin_amdgcn_wmma_f32_16x16x4_f32(false, a, false, bones,
                                                   (short)0, acc1, false, false);
      acca = __builtin_amdgcn_wmma_f32_16x16x4_f32(false, a, false, balt,
                                                   (short)0, acca, false, false);
    }
    // D[m][n] identical across n (B columns identical) -> sum over all lanes/VGPRs = 16 * sum_m
    float lb = 0.0f;
    #pragma unroll
    for (int r = 0; r < 8; ++r) lb += acc1[r] * acc1[r] + acca[r] * acca[r];
    totB += lb;
#else
    // Exact VALU fallback: lane l and lane l+16 each hold half of column l&15.
    float s1p = 0.0f, sap = 0.0f;
    #pragma unroll
    for (int k = 0; k < 32; ++k) {
      const v2f a = *(const v2f*)(cptr + 4 * k);
      totA += a.x * a.x + a.y * a.y;
      s1p += a.x + a.y;     // z even + z odd
      sap += a.x - a.y;     // (+1)*even + (-1)*odd
    }
    const float s1 = s1p + __shfl_xor(s1p, 16, 32);
    const float sa = sap + __shfl_xor(sap, 16, 32);
    totB += (lane < 16) ? (s1 * s1 + sa * sa) : 0.0f;
#endif
    __syncthreads();   // protect LDS tile before next stage overwrites it
  }

#if USE_WMMA_F32X4
  totB *= (1.0f / 16.0f);   // undo 16x replication across D columns
#endif

  // wave32 butterfly reduce
  #pragma unroll
  for (int off = 16; off > 0; off >>= 1) {
    totA += __shfl_xor(totA, off, 32);
    totB += __shfl_xor(totB, off, 32);
  }
  if (lane == 0) { wA[wave] = totA; wB[wave] = totB; }
  __syncthreads();
  if (threadIdx.x == 0) {
    float A = 0.0f, B = 0.0f;
    #pragma unroll
    for (int w = 0; w < WAVES; ++w) { A += wA[w]; B += wB[w]; }
    part[2 * blockIdx.x + 0] = A;
    part[2 * blockIdx.x + 1] = B;
  }
}

__global__ __launch_bounds__(256)
void spectral_finalize_kernel(const float* __restrict__ part, int nblocks,
                              float* __restrict__ out) {
  __shared__ float sA[256], sB[256];
  float a = 0.0f, b = 0.0f;
  for (int i = threadIdx.x; i < nblocks; i += 256) {
    a += part[2 * i + 0];
    b += part[2 * i + 1];
  }
  sA[threadIdx.x] = a; sB[threadIdx.x] = b;
  __syncthreads();
  for (int s = 128; s > 0; s >>= 1) {
    if ((int)threadIdx.x < s) { sA[threadIdx.x] += sA[threadIdx.x + s];
                                sB[threadIdx.x] += sB[threadIdx.x + s]; }
    __syncthreads();
  }
  if (threadIdx.x == 0) {
    const float A = sA[0];                 // sum x^2
    const float B = sB[0];                 // sum (s1^2 + sa^2)
    const float E = 0.5f * (2097152.0f * A + 16384.0f * B);   // 128^3, 128^2
    const float spectrum1 = E / (1064960.0f * 4.0f);          // count=128*128*65, b=4
    // constant decay term, computed exactly as the reference does
    const float slope = -5.0f / 3.0f;
    const float denom = __powf(17.0f, slope) + 1e-8f;
    float c = 0.0f;
    for (int k = 17; k < 64; ++k) {
      const float en = __powf((float)k, slope) / denom;
      c += en * en;
    }
    c *= (1.0f / 47.0f);
    out[0] = 0.01f * c + 0.001f * (spectrum1 / 63.0f);
  }
}

extern "C" void kernel_launch(void* const* d_in, const int* in_sizes, int n_in,
                              void* d_out, int out_size, void* d_ws, size_t ws_size,
                              hipStream_t stream) {
  const float* x = (const float*)d_in[0];
  float* out = (float*)d_out;
  float* part = (float*)d_ws;   // NBLK_A * 2 floats = 4 KB
  (void)in_sizes; (void)n_in; (void)out_size; (void)ws_size;

  spectral_partial_kernel<<<NBLK_A, BLOCK_A, 0, stream>>>(x, part);
  spectral_finalize_kernel<<<1, 256, 0, stream>>>(part, NBLK_A, out);
}